// Noise2Void_73229192397062
// MI455X (gfx1250) — compile-verified
//
#include <hip/hip_runtime.h>
#include <hip/hip_bf16.h>
#include <stdint.h>

typedef __attribute__((ext_vector_type(2))) float v2f;
typedef __attribute__((ext_vector_type(8))) float v8f;

// ---------------------------------------------------------------------------
// JAX-faithful Threefry-2x32 (20 rounds), matching jax._src.prng.threefry2x32
// ---------------------------------------------------------------------------
__device__ __forceinline__ uint32_t rotl32(uint32_t x, uint32_t r) {
  return (x << r) | (x >> (32u - r));
}

__device__ __forceinline__ void threefry2x32(uint32_t k0, uint32_t k1,
                                             uint32_t c0, uint32_t c1,
                                             uint32_t& o0, uint32_t& o1) {
  const uint32_t ks2 = k0 ^ k1 ^ 0x1BD11BDAu;
  uint32_t x0 = c0 + k0;
  uint32_t x1 = c1 + k1;
#define TF_ROUND(r) { x0 += x1; x1 = rotl32(x1, r); x1 ^= x0; }
  TF_ROUND(13) TF_ROUND(15) TF_ROUND(26) TF_ROUND(6)
  x0 += k1;  x1 += ks2 + 1u;
  TF_ROUND(17) TF_ROUND(29) TF_ROUND(16) TF_ROUND(24)
  x0 += ks2; x1 += k0 + 2u;
  TF_ROUND(13) TF_ROUND(15) TF_ROUND(26) TF_ROUND(6)
  x0 += k0;  x1 += k1 + 3u;
  TF_ROUND(17) TF_ROUND(29) TF_ROUND(16) TF_ROUND(24)
  x0 += k1;  x1 += ks2 + 4u;
  TF_ROUND(13) TF_ROUND(15) TF_ROUND(26) TF_ROUND(6)
  x0 += ks2; x1 += k0 + 5u;
#undef TF_ROUND
  o0 = x0; o1 = x1;
}

// bits[idx] of jax _random_bits(key, 32, size) with size = 2*half:
// counter pair (i, i+half) -> (bits[i], bits[i+half])
__device__ __forceinline__ uint32_t jax_random_bits(uint32_t k0, uint32_t k1,
                                                    uint32_t idx, uint32_t half) {
  uint32_t o0, o1;
  if (idx < half) { threefry2x32(k0, k1, idx, idx + half, o0, o1); return o0; }
  threefry2x32(k0, k1, idx - half, idx, o0, o1);
  return o1;
}

// ---------------------------------------------------------------------------
// Kernel 1: one thread per hot pixel (N*C*K = 24576). Regenerates the exact
// JAX randomness, gathers the two pixels, accumulates (repl-orig)^2, block-
// reduces via wave32 shuffles + LDS, writes one partial per block to d_ws.
// ---------------------------------------------------------------------------
__global__ __launch_bounds__(256)
void n2v_err_kernel(const float* __restrict__ x, float* __restrict__ partials) {
  const uint32_t g = blockIdx.x * 256u + threadIdx.x;   // 0..24575

  // jax.random.key(42) == (0,42); split -> k1 (randint), k2 (uniform).
  // All-constant args: the compiler folds these two calls to immediates.
  uint32_t k1a, k1b, k2a, k2b, t0, t1;
  threefry2x32(0u, 42u, 0u, 2u, t0, t1); k1a = t0; k2a = t1;
  threefry2x32(0u, 42u, 1u, 3u, t0, t1); k1b = t0; k2b = t1;

  // randint(k1, (N,C,16,16,2), 0, 32): flat bits array of 49152, span=32 -> &31
  const uint32_t offy = jax_random_bits(k1a, k1b, 2u * g,      24576u) & 31u;
  const uint32_t offx = jax_random_bits(k1a, k1b, 2u * g + 1u, 24576u) & 31u;
  // uniform(k2, (N,C,256)): flat bits array of 24576
  const uint32_t ub = jax_random_bits(k2a, k2b, g, 12288u);
  const float u = __uint_as_float((ub >> 9) | 0x3F800000u) - 1.0f;

  const int nc = (int)(g >> 8);                 // n*3 + c
  const int k  = (int)(g & 255u);               // by*16 + bx
  const int hy = (k >> 4) * 32 + (int)offy;
  const int hx = (k & 15) * 32 + (int)offx;

  // Faithful ROI quirks: roimax clipped to 511 (shape-1), center rejection
  // only for flat rc index 2*sh1+2 when sh0>2 && sh1>2.
  const int rmin0 = max(hy - 2, 0), rmax0 = min(hy + 3, 511);
  const int rmin1 = max(hx - 2, 0), rmax1 = min(hx + 3, 511);
  const int sh0 = rmax0 - rmin0, sh1 = rmax1 - rmin1;
  const bool has_center = (sh0 > 2) && (sh1 > 2);
  const int m = sh0 * sh1 - (has_center ? 1 : 0);
  int ui = (int)floorf(u * (float)m);
  ui = min(ui, m - 1);
  const int cflat = 2 * sh1 + 2;
  if (has_center && ui >= cflat) ui += 1;
  const int ry = rmin0 + ui / sh1;
  const int rx = rmin1 + ui % sh1;

  const int base = nc * (512 * 512);
  const float orig = x[base + hy * 512 + hx];
  const float repl = x[base + ry * 512 + rx];
  const float d = repl - orig;
  float e = d * d;

  // wave32 butterfly reduce, then 8 wave results via LDS
  #pragma unroll
  for (int off = 16; off > 0; off >>= 1) e += __shfl_xor(e, off, 32);
  __shared__ float red[8];
  if ((threadIdx.x & 31u) == 0u) red[threadIdx.x >> 5] = e;
  __syncthreads();
  if (threadIdx.x == 0) {
    float s = 0.0f;
    #pragma unroll
    for (int i = 0; i < 8; ++i) s += red[i];
    partials[blockIdx.x] = s;
  }
}

// ---------------------------------------------------------------------------
// Kernel 2: single wave32. Reduce the 96 block partials through the matrix
// pipe: two accumulating V_WMMA_F32_16X16X4_F32 with B = ones(4x16).
// A layout (16x4 f32): lane m VGPR0/1 -> A[m,0]/A[m,1]; lane m+16 -> A[m,2]/A[m,3].
// With B=ones, D[m,n] = row-sum of A (every column identical), so after two
// accumulating calls each D element chain covers 6 partials + 2 zero pads; summing a lane's
// 8 C VGPRs gives rows 0-7 (lane 0) / 8-15 (lane 16); one shfl_xor finishes.
// ---------------------------------------------------------------------------
__global__ __launch_bounds__(32)
void n2v_wmma_reduce(const float* __restrict__ partials, float* __restrict__ out) {
  const int lane = threadIdx.x;            // exactly one wave32, EXEC all ones
  const float a0 = partials[lane];         // slots 0..31
  const float a1 = partials[lane + 32];    // slots 32..63
  const float a2 = partials[lane + 64];    // slots 64..95
  v2f A0 = {a0, a1};
  v2f A1 = {a2, 0.0f};                     // zero-pad slots 96..127
  v2f B  = {1.0f, 1.0f};                   // ones matrix
  v8f c  = {};
  c = __builtin_amdgcn_wmma_f32_16x16x4_f32(false, A0, false, B, (short)0, c,
                                            false, false);
  c = __builtin_amdgcn_wmma_f32_16x16x4_f32(false, A1, false, B, (short)0, c,
                                            false, false);
  float s = c[0] + c[1] + c[2] + c[3] + c[4] + c[5] + c[6] + c[7];
  s += __shfl_xor(s, 16, 32);              // lane0: rows0-7 + rows8-15 = total
  if (lane == 0) out[0] = s * (1.0f / 24576.0f);
}

// ---------------------------------------------------------------------------
extern "C" void kernel_launch(void* const* d_in, const int* in_sizes, int n_in,
                              void* d_out, int out_size, void* d_ws, size_t ws_size,
                              hipStream_t stream) {
  (void)in_sizes; (void)n_in; (void)out_size; (void)ws_size;
  const float* x = (const float*)d_in[0];       // (32,3,512,512) fp32
  float* out = (float*)d_out;                   // scalar fp32
  float* partials = (float*)d_ws;               // 96 floats, rewritten each call

  n2v_err_kernel<<<96, 256, 0, stream>>>(x, partials);       // 96*256 = 24576
  n2v_wmma_reduce<<<1, 32, 0, stream>>>(partials, out);
}